// SparseCenterHead_85126251807278
// MI455X (gfx1250) — compile-verified
//
#include <hip/hip_runtime.h>
#include <hip/hip_bf16.h>
#include <cstdint>

// CDNA5 / gfx1250: wave32, WMMA 16x16x4 f32.
typedef float v2f __attribute__((ext_vector_type(2)));
typedef float v8f __attribute__((ext_vector_type(8)));

#define K_OBJ       500
#define K_PAD       512          // 32 tiles of 16 rows
#define N_VOXELS    150000       // 9375 tiles of 16 cols (exact)
#define N_TILES     (N_VOXELS / 16)
#define NUM_CLASSES 3
#define MIN_RAD     2
#define OVERLAP_F   0.1f
#define COORD_SCALE 1.25f        // 1/(voxel 0.1 * stride 8)
#define COORD_OFF   75.2f        // -pc_range[0]
#define COORD_MAX   187.5f       // FMAP - 0.5
#define LOG2E       1.44269504088896340736f

#define CHUNKS      125          // argmin column chunks
#define TILES_PER_CHUNK (N_TILES / CHUNKS)   // 75

// ---------------------------------------------------------------- pass 1
// Per-object integer-center A values; init argmin keys.
__global__ void sch_prep(const float* __restrict__ gt,
                         float2* __restrict__ cA,
                         unsigned long long* __restrict__ keyArr) {
    int k = blockIdx.x * blockDim.x + threadIdx.x;
    if (k >= K_PAD) return;
    keyArr[k] = ~0ull;
    if (k >= K_OBJ) { cA[k] = make_float2(0.f, 0.f); return; }
    const float* b = gt + 8 * k;
    float cx = fminf(fmaxf((b[0] + COORD_OFF) * COORD_SCALE, 0.f), COORD_MAX);
    float cy = fminf(fmaxf((b[1] + COORD_OFF) * COORD_SCALE, 0.f), COORD_MAX);
    cA[k] = make_float2((float)(int)cx, (float)(int)cy);   // center_int as float
}

// ---------------------------------------------------------------- pass 2
// Per-object min/argmin(dist) over all voxels, via WMMA dist tiles and
// packed u64 keys (dist<<18 | idx) reduced with global_atomic_min_u64.
__global__ void sch_argmin(const int* __restrict__ sidx,
                           const float2* __restrict__ cA,
                           unsigned long long* __restrict__ keyArr) {
    const int lane = threadIdx.x;        // wave32: one wave per block
    const int half = lane >> 4;
    const int l15  = lane & 15;
    const int objBase = blockIdx.y * 16;

    float2 c = cA[objBase + l15];
    v2f a;
    a[0] = half ? (c.x * c.x + c.y * c.y) : c.x;
    a[1] = half ? 1.0f                    : c.y;

    unsigned long long best[8];
#pragma unroll
    for (int v = 0; v < 8; ++v) best[v] = ~0ull;

    const int t0 = blockIdx.x * TILES_PER_CHUNK;
    const int t1 = t0 + TILES_PER_CHUNK;
    for (int t = t0; t < t1; ++t) {
        const int n = t * 16 + l15;
        const float fx = (float)sidx[2 * n];
        const float fy = (float)sidx[2 * n + 1];
        v2f bb;
        bb[0] = half ? 1.0f                    : -2.f * fx;
        bb[1] = half ? (fx * fx + fy * fy)     : -2.f * fy;
        v8f c0 = {};
        v8f d = __builtin_amdgcn_wmma_f32_16x16x4_f32(
            false, a, false, bb, (short)0, c0, false, false);
        const unsigned long long idx = (unsigned long long)(unsigned)n;
#pragma unroll
        for (int v = 0; v < 8; ++v) {
            unsigned di = __float2uint_rn(d[v]);   // dist is exactly integral
            unsigned long long key = ((unsigned long long)di << 18) | idx;
            best[v] = best[v] < key ? best[v] : key;
        }
    }
#pragma unroll
    for (int v = 0; v < 8; ++v) {
        unsigned long long m = best[v];
        for (int off = 1; off < 16; off <<= 1) {
            unsigned long long o = __shfl_xor(m, off, 32);
            m = m < o ? m : o;
        }
        if (l15 == 0)
            atomicMin(&keyArr[objBase + 8 * half + v], m);
    }
}

// ---------------------------------------------------------------- pass 3
// O(K): gaussian radius/sigma, normalization factor, ret_boxes/inds/valid,
// and packed per-row params for the heatmap pass.
__global__ void sch_finalize(const float* __restrict__ gt,
                             const int* __restrict__ sidx,
                             const unsigned long long* __restrict__ keyArr,
                             float4* __restrict__ objp,
                             float* __restrict__ out) {
    int k = blockIdx.x * blockDim.x + threadIdx.x;
    if (k >= K_PAD) return;
    if (k >= K_OBJ) { objp[k] = make_float4(0.f, 0.f, -1.f, 0.f); return; }

    const float* b = gt + 8 * k;
    const bool valid = (b[3] > 0.f) && (b[4] > 0.f);

    float cx = fminf(fmaxf((b[0] + COORD_OFF) * COORD_SCALE, 0.f), COORD_MAX);
    float cy = fminf(fmaxf((b[1] + COORD_OFF) * COORD_SCALE, 0.f), COORD_MAX);

    // gaussian_radius(height=dx, width=dy, OVERLAP)
    const float h = b[3] * COORD_SCALE;
    const float w = b[4] * COORD_SCALE;
    const float b1 = h + w;
    const float c1 = w * h * (1.f - OVERLAP_F) / (1.f + OVERLAP_F);
    const float r1 = (b1 + sqrtf(b1 * b1 - 4.f * c1)) * 0.5f;
    const float b2 = 2.f * (h + w);
    const float c2 = (1.f - OVERLAP_F) * w * h;
    const float r2 = (b2 + sqrtf(b2 * b2 - 16.f * c2)) * 0.125f;
    const float a3 = 4.f * OVERLAP_F;
    const float b3 = -2.f * OVERLAP_F * (h + w);
    const float c3 = (OVERLAP_F - 1.f) * w * h;
    const float r3 = (-b3 + sqrtf(b3 * b3 - 4.f * a3 * c3)) / (2.f * a3);
    int rad = (int)fminf(fminf(r1, r2), r3);       // trunc, matches astype(int32)
    rad = rad > MIN_RAD ? rad : MIN_RAD;
    const float sigma  = (2.f * (float)rad + 1.f) / 6.f;
    const float inv2s2 = 1.f / (2.f * sigma * sigma);

    const unsigned long long key = keyArr[k];
    const unsigned mind = (unsigned)(key >> 18);
    const unsigned idx  = (unsigned)(key & 0x3FFFFull);
    const float maxg   = __expf(-(float)mind * inv2s2);
    const float invden = 1.f / fmaxf(maxg, 1e-6f);

    objp[k] = make_float4(-inv2s2 * LOG2E, invden,
                          (float)((int)b[7] - 1), valid ? 1.f : 0.f);

    // --- outputs: ret_boxes [500,8], inds [500], valid [500]
    float* retb = out + NUM_CLASSES * N_VOXELS;
    float* inds = retb + K_OBJ * 8;
    float* valo = inds + K_OBJ;
    const float nx = (float)sidx[2 * idx];
    const float ny = (float)sidx[2 * idx + 1];
    float rb[8];
    rb[0] = cx - nx;  rb[1] = cy - ny;  rb[2] = b[2];
    rb[3] = logf(valid ? b[3] : 1.f);
    rb[4] = logf(valid ? b[4] : 1.f);
    rb[5] = logf(valid ? b[5] : 1.f);
    rb[6] = cosf(b[6]); rb[7] = sinf(b[6]);
#pragma unroll
    for (int j = 0; j < 8; ++j) retb[8 * k + j] = valid ? rb[j] : 0.f;
    inds[k] = (float)idx;
    valo[k] = valid ? 1.f : 0.f;
}

// ---------------------------------------------------------------- pass 4
// Heatmap: one wave per 16-voxel tile; WMMA dist vs all 32 object tiles,
// exp2 + per-class masked running max, shfl_xor(16) to merge wave halves.
__global__ void sch_heatmap(const int* __restrict__ sidx,
                            const float2* __restrict__ cA,
                            const float4* __restrict__ objp,
                            float* __restrict__ hm) {
    const int lane = threadIdx.x;
    const int half = lane >> 4;
    const int l15  = lane & 15;
    const int n    = blockIdx.x * 16 + l15;

    const float fx = (float)sidx[2 * n];
    const float fy = (float)sidx[2 * n + 1];
    v2f bb;
    bb[0] = half ? 1.0f                : -2.f * fx;
    bb[1] = half ? (fx * fx + fy * fy) : -2.f * fy;

    float hm0 = 0.f, hm1 = 0.f, hm2 = 0.f;
    for (int ot = 0; ot < K_PAD / 16; ++ot) {
        float2 c = cA[ot * 16 + l15];
        v2f a;
        a[0] = half ? (c.x * c.x + c.y * c.y) : c.x;
        a[1] = half ? 1.0f                    : c.y;
        v8f c0 = {};
        v8f d = __builtin_amdgcn_wmma_f32_16x16x4_f32(
            false, a, false, bb, (short)0, c0, false, false);
        const int rbase = ot * 16 + 8 * half;
#pragma unroll
        for (int v = 0; v < 8; ++v) {
            float4 p = objp[rbase + v];
            float g = exp2f(d[v] * p.x) * p.y * p.w;   // normalized gauss, 0 if invalid
            hm0 = (p.z == 0.f) ? fmaxf(hm0, g) : hm0;
            hm1 = (p.z == 1.f) ? fmaxf(hm1, g) : hm1;
            hm2 = (p.z == 2.f) ? fmaxf(hm2, g) : hm2;
        }
    }
    hm0 = fmaxf(hm0, __shfl_xor(hm0, 16, 32));
    hm1 = fmaxf(hm1, __shfl_xor(hm1, 16, 32));
    hm2 = fmaxf(hm2, __shfl_xor(hm2, 16, 32));
    if (half == 0) {
        hm[0 * N_VOXELS + n] = hm0;
        hm[1 * N_VOXELS + n] = hm1;
        hm[2 * N_VOXELS + n] = hm2;
    }
}

// ---------------------------------------------------------------- launch
extern "C" void kernel_launch(void* const* d_in, const int* in_sizes, int n_in,
                              void* d_out, int out_size, void* d_ws, size_t ws_size,
                              hipStream_t stream) {
    const float* gt   = (const float*)d_in[0];   // [500, 8]
    const int*   sidx = (const int*)d_in[1];     // [150000, 2]
    float* out = (float*)d_out;

    // workspace layout
    float2* cA = (float2*)d_ws;                                        // 4 KB
    unsigned long long* keyArr =
        (unsigned long long*)((char*)d_ws + 4096);                     // 4 KB
    float4* objp = (float4*)((char*)d_ws + 8192);                      // 8 KB

    sch_prep<<<K_PAD / 128, 128, 0, stream>>>(gt, cA, keyArr);
    sch_argmin<<<dim3(CHUNKS, K_PAD / 16), 32, 0, stream>>>(sidx, cA, keyArr);
    sch_finalize<<<K_PAD / 128, 128, 0, stream>>>(gt, sidx, keyArr, objp, out);
    sch_heatmap<<<N_TILES, 32, 0, stream>>>(sidx, cA, objp, out);
}